// RelationalAttentionLayer_3504693314217
// MI455X (gfx1250) — compile-verified
//
#include <hip/hip_runtime.h>
#include <hip/hip_bf16.h>

// ---------------------------------------------------------------------------
// RelationalAttentionLayer for MI455X (gfx1250)
//
// Decomposition: c_e = P[row] + Q[col] + (R+b1)[type], with
//   P = h @ W1[:, 0:100]^T   [N,200]
//   Q = h @ W1[:, 100:200]^T [N,200]   (stored interleaved as PQ [N,400])
//   R = g @ W1[:, 200:300]^T + b1 [237,200]
// Attention logit is linear in c => per-node scalars pA,pB and per-rel pG.
// Everything hot (PQ 80MB, out 40MB, edge scalars) fits MI455X's 192MB L2.
// ---------------------------------------------------------------------------

#define HEADS    2
#define OUTSZ    100
#define IN_H     100
#define IN_G     100
#define N_NODES  50000
#define N_EDGES  500000
#define N_REL    237
#define FANIN    300          // 2*IN_H + IN_G
#define HO       (HEADS*OUTSZ) // 200
#define PQW      (2*HO)        // 400

typedef __attribute__((ext_vector_type(2))) float v2f;
typedef __attribute__((ext_vector_type(8))) float v8f;

// ---------------------------------------------------------------------------
// Zero-fill (graph-capture safe; re-run every call for determinism)
// ---------------------------------------------------------------------------
__global__ void zero_kernel(float* __restrict__ p, unsigned n) {
    unsigned i = blockIdx.x * blockDim.x + threadIdx.x;
    if (i < n) p[i] = 0.0f;
}

// ---------------------------------------------------------------------------
// PQ = h @ [W1_h^T | W1_t^T]  -> [N_NODES, 400], fp32 WMMA 16x16x4.
// One wave per 16x16 output tile. K=100 = 25 steps of 4.
//
// 32-bit A layout (16x4):  lanes 0-15 M=0..15, VGPR0={K0,K2}, VGPR1={K1,K3}
//   -> lane: m = lane&15, kb = k0 + (lane>>4)*2, a = {h[m][kb], h[m][kb+1]}
// 32-bit B layout (4x16):  mirrored: n = lane&15, same kb.
// 32-bit C layout (16x16): lane n = lane&15, VGPR r -> M = r + 8*(lane>>4)
// ---------------------------------------------------------------------------
__global__ void gemm_pq_kernel(const float* __restrict__ h,
                               const float* __restrict__ W1,
                               float* __restrict__ PQ) {
    const int lane  = threadIdx.x & 31;
    const int wid   = blockIdx.x * (blockDim.x >> 5) + (threadIdx.x >> 5);
    const int NT_N  = PQW / 16;               // 25
    const int NT_M  = N_NODES / 16;           // 3125 (exact)
    if (wid >= NT_M * NT_N) return;           // whole-wave uniform exit (EXEC all 1s in WMMA)
    const int tm    = wid / NT_N;
    const int tn    = wid % NT_N;
    const int half  = lane >> 4;              // 0 or 1
    const int l15   = lane & 15;
    const int m     = tm * 16 + l15;          // A row for this lane
    const int n     = tn * 16 + l15;          // B col for this lane

    // B(k, n) = n<200 ? W1[n][k] : W1[n-200][100+k]
    const float* __restrict__ wp = (n < HO) ? (W1 + n * FANIN)
                                            : (W1 + (n - HO) * FANIN + IN_H);
    const float* __restrict__ hp = h + m * IN_H;

    v8f acc = {0.f, 0.f, 0.f, 0.f, 0.f, 0.f, 0.f, 0.f};
#pragma unroll
    for (int ks = 0; ks < IN_H / 4; ++ks) {
        const int kb = ks * 4 + half * 2;
        v2f a, b;
        a.x = hp[kb];  a.y = hp[kb + 1];
        b.x = wp[kb];  b.y = wp[kb + 1];
        // 8 args: (neg_a, A, neg_b, B, c_mod, C, reuse_a, reuse_b)
        acc = __builtin_amdgcn_wmma_f32_16x16x4_f32(
            false, a, false, b, (short)0, acc, false, false);
    }

    const int mbase = tm * 16 + half * 8;
#pragma unroll
    for (int r = 0; r < 8; ++r) {
        PQ[(mbase + r) * PQW + n] = acc[r];
    }
}

// ---------------------------------------------------------------------------
// R[rel][j] = b1[j] + sum_k g[rel][k] * W1[j][200+k]   (237x200, tiny)
// ---------------------------------------------------------------------------
__global__ void rel_kernel(const float* __restrict__ g,
                           const float* __restrict__ W1,
                           const float* __restrict__ b1,
                           float* __restrict__ R) {
    int tid = blockIdx.x * blockDim.x + threadIdx.x;
    if (tid >= N_REL * HO) return;
    int rel = tid / HO, j = tid % HO;
    const float* gp = g + rel * IN_G;
    const float* wp = W1 + j * FANIN + 2 * IN_H;
    float acc = b1[j];
#pragma unroll 4
    for (int k = 0; k < IN_G; ++k) acc += gp[k] * wp[k];
    R[tid] = acc;
}

// ---------------------------------------------------------------------------
// pA[n][hd] = w_att[hd] . P[n][hd]   ;   pB[n][hd] = w_att[hd] . Q[n][hd]
// ---------------------------------------------------------------------------
__global__ void node_att_kernel(const float* __restrict__ PQ,
                                const float* __restrict__ w_att,
                                float* __restrict__ pA,
                                float* __restrict__ pB) {
    int tid = blockIdx.x * blockDim.x + threadIdx.x;
    if (tid >= N_NODES * HEADS) return;
    int node = tid >> 1, hd = tid & 1;
    const float* p = PQ + (size_t)node * PQW + hd * OUTSZ;
    const float* q = p + HO;
    const float* w = w_att + hd * OUTSZ;
    float a = 0.f, b = 0.f;
#pragma unroll 4
    for (int o = 0; o < OUTSZ; ++o) { float wv = w[o]; a += wv * p[o]; b += wv * q[o]; }
    pA[tid] = a;
    pB[tid] = b;
}

// pG[rel][hd] = w_att[hd] . R[rel][hd]   (bias folded into R)
__global__ void relatt_kernel(const float* __restrict__ R,
                              const float* __restrict__ w_att,
                              float* __restrict__ pG) {
    int tid = blockIdx.x * blockDim.x + threadIdx.x;
    if (tid >= N_REL * HEADS) return;
    int rel = tid >> 1, hd = tid & 1;
    const float* rp = R + rel * HO + hd * OUTSZ;
    const float* w  = w_att + hd * OUTSZ;
    float a = 0.f;
#pragma unroll 4
    for (int o = 0; o < OUTSZ; ++o) a += w[o] * rp[o];
    pG[tid] = a;
}

// ---------------------------------------------------------------------------
// Edge pass 1: logits -> exp -> row_sum (segment sum over source node)
// ---------------------------------------------------------------------------
__global__ void edge_logits_kernel(const int* __restrict__ eidx,
                                   const int* __restrict__ etype,
                                   const float* __restrict__ pA,
                                   const float* __restrict__ pB,
                                   const float* __restrict__ pG,
                                   float* __restrict__ eV,
                                   float* __restrict__ row_sum) {
    int e = blockIdx.x * blockDim.x + threadIdx.x;
    if (e >= N_EDGES) return;
    int r = eidx[e];
    int c = eidx[N_EDGES + e];
    int t = etype[e];
#pragma unroll
    for (int hd = 0; hd < HEADS; ++hd) {
        float s = pA[r * 2 + hd] + pB[c * 2 + hd] + pG[t * 2 + hd];
        float l = (s > 0.f) ? s : s * 0.01f;       // leaky_relu(0.01)
        float ex = __expf(l);
        eV[e * 2 + hd] = ex;
        atomicAdd(&row_sum[r * 2 + hd], ex);
    }
}

// ---------------------------------------------------------------------------
// Edge pass 2: out[col] += alpha * (P[row] + Q[col] + R[type])
// One thread per (edge, hd*100+o): 100M threads, L2-resident gathers,
// global_atomic_add_f32 scatter.
// ---------------------------------------------------------------------------
__global__ void edge_scatter_kernel(const int* __restrict__ eidx,
                                    const int* __restrict__ etype,
                                    const float* __restrict__ PQ,
                                    const float* __restrict__ R,
                                    const float* __restrict__ eV,
                                    const float* __restrict__ row_sum,
                                    float* __restrict__ out) {
    unsigned gid = blockIdx.x * blockDim.x + threadIdx.x;
    const unsigned total = (unsigned)N_EDGES * (unsigned)HO;   // 100,000,000
    if (gid >= total) return;
    unsigned e  = gid / (unsigned)HO;
    unsigned t  = gid - e * (unsigned)HO;     // t = hd*100 + o
    unsigned hd = t / (unsigned)OUTSZ;
    int r  = eidx[e];
    int c  = eidx[N_EDGES + e];
    int ty = etype[e];
    float alpha = eV[e * 2 + hd] / row_sum[(unsigned)r * 2 + hd];
    float val = PQ[(size_t)r * PQW + t]
              + PQ[(size_t)c * PQW + HO + t]
              + R[(size_t)ty * HO + t];
    atomicAdd(&out[(size_t)c * HO + t], alpha * val);
}

// ---------------------------------------------------------------------------
// Host-side launch
// ---------------------------------------------------------------------------
extern "C" void kernel_launch(void* const* d_in, const int* in_sizes, int n_in,
                              void* d_out, int out_size, void* d_ws, size_t ws_size,
                              hipStream_t stream) {
    // setup_inputs order: x, h, g, edge_idx, edge_type, W1, b1, w_att
    const float* h     = (const float*)d_in[1];
    const float* g     = (const float*)d_in[2];
    const int*   eidx  = (const int*)d_in[3];
    const int*   etype = (const int*)d_in[4];
    const float* W1    = (const float*)d_in[5];
    const float* b1    = (const float*)d_in[6];
    const float* w_att = (const float*)d_in[7];
    float* out = (float*)d_out;

    // Workspace carve-up (256B aligned), total ~85.4 MB
    char* ws = (char*)d_ws;
    size_t off = 0;
    auto carve = [&](size_t bytes) -> float* {
        float* p = (float*)(ws + off);
        off += (bytes + 255) & ~(size_t)255;
        return p;
    };
    float* PQ      = carve((size_t)N_NODES * PQW * sizeof(float));   // 80 MB
    float* R       = carve((size_t)N_REL * HO * sizeof(float));
    float* pA      = carve((size_t)N_NODES * HEADS * sizeof(float));
    float* pB      = carve((size_t)N_NODES * HEADS * sizeof(float));
    float* pG      = carve((size_t)N_REL * HEADS * sizeof(float));
    float* eV      = carve((size_t)N_EDGES * HEADS * sizeof(float)); // 4 MB
    float* row_sum = carve((size_t)N_NODES * HEADS * sizeof(float));
    (void)ws_size;

    const int B = 256;

    // 0) zero out + row_sum (every call; buffers persist across graph replays)
    {
        unsigned n_out = (unsigned)N_NODES * HO;          // 10,000,000
        zero_kernel<<<(n_out + B - 1) / B, B, 0, stream>>>(out, n_out);
        unsigned n_rs = (unsigned)N_NODES * HEADS;
        zero_kernel<<<(n_rs + B - 1) / B, B, 0, stream>>>(row_sum, n_rs);
    }

    // 1) WMMA GEMM: PQ = h @ [W1_h^T | W1_t^T]
    {
        const int waves_per_blk = B / 32;                 // 8
        const int n_waves = (N_NODES / 16) * (PQW / 16);  // 78,125 tiles
        const int blocks = (n_waves + waves_per_blk - 1) / waves_per_blk;
        gemm_pq_kernel<<<blocks, B, 0, stream>>>(h, W1, PQ);
    }

    // 2) R = g @ W1_g^T + b1
    rel_kernel<<<(N_REL * HO + B - 1) / B, B, 0, stream>>>(g, W1, b1, R);

    // 3) per-node / per-rel attention scalars
    node_att_kernel<<<(N_NODES * HEADS + B - 1) / B, B, 0, stream>>>(PQ, w_att, pA, pB);
    relatt_kernel<<<(N_REL * HEADS + B - 1) / B, B, 0, stream>>>(R, w_att, pG);

    // 4) edge logits + segment sum over source node
    edge_logits_kernel<<<(N_EDGES + B - 1) / B, B, 0, stream>>>(
        eidx, etype, pA, pB, pG, eV, row_sum);

    // 5) normalized weighted scatter into destination nodes
    {
        unsigned total = (unsigned)N_EDGES * (unsigned)HO;  // 100M
        edge_scatter_kernel<<<(total + B - 1) / B, B, 0, stream>>>(
            eidx, etype, PQ, R, eV, row_sum, out);
    }
}